// ConvChunk2d_22780506538661
// MI455X (gfx1250) — compile-verified
//
#include <hip/hip_runtime.h>
#include <hip/hip_bf16.h>
#include <stdint.h>

// ConvChunk2d: pure gather / im2col-with-channel-scramble.
//   out[n,y,x,c,ky,kx] = xpad[n, ((ky*3+kx)*64+c)/9, y+ky-1, x+kx-1]
// N=8, C=64, H=W=128, K=3, S=1, P=1. Output (N*H*W, 64, 3, 3) f32.

#define CIN   64
#define HH    128
#define WW    128
#define TX    64            // pixels per workgroup (one row segment)
#define COLS  (TX + 2)      // 66 halo columns
#define ST    67            // LDS row stride (spreads 64 banks)
#define NTILE (CIN * 3 * COLS)   // 12672 tile elements
#define NTHR  288           // 9 waves; 144 float4 slots x 2 pixel-halves
#define NSLOT 144           // float4 slots per pixel row (576/4)
#define HPIX  32            // pixels per half

typedef float v4f __attribute__((ext_vector_type(4)));

#if defined(__HIP_DEVICE_COMPILE__) && __has_builtin(__builtin_amdgcn_global_load_async_to_lds_b32)
#define USE_ASYNC_COPY 1
#else
#define USE_ASYNC_COPY 0
#endif

#define GAS __attribute__((address_space(1)))
#define LAS __attribute__((address_space(3)))

__device__ __forceinline__ int gather_base(int e)
{
    // e = c*9 + ky*3 + kx ; source channel c_src = ((ky*3+kx)*64 + c) / 9
    int c  = e / 9;
    int r  = e - 9 * c;
    int ky = r / 3;
    int kx = r - 3 * ky;
    int cs = (r * 64 + c) / 9;
    return (cs * 3 + ky) * ST + kx;
}

__global__ __launch_bounds__(NTHR) void convchunk_gather_kernel(
    const float* __restrict__ x, float* __restrict__ out)
{
    __shared__ float tile[CIN * 3 * ST];   // 64*3*67*4 = 51456 B

    const int tid = threadIdx.x;
    const int x0  = blockIdx.x * TX;       // 0 or 64
    const int y   = blockIdx.y;            // 0..127
    const int n   = blockIdx.z;            // 0..7

    const float* __restrict__ xn = x + (size_t)n * CIN * HH * WW;

    // ---- Stage halo tile (all 64 channels, rows y-1..y+1, cols x0-1..x0+64) ----
    for (int i = tid; i < NTILE; i += NTHR) {
        int c   = i / (3 * COLS);
        int r2  = i - c * (3 * COLS);
        int ry  = r2 / COLS;
        int cx  = r2 - ry * COLS;
        int gy  = y  + ry - 1;
        int gx  = x0 + cx - 1;
        int lofs = (c * 3 + ry) * ST + cx;
        if ((unsigned)gy < (unsigned)HH && (unsigned)gx < (unsigned)WW) {
            const float* src = xn + ((size_t)c * HH + gy) * WW + gx;
#if USE_ASYNC_COPY
            // CDNA5 async global->LDS DMA (ASYNCcnt), per-lane addresses.
            __builtin_amdgcn_global_load_async_to_lds_b32(
                (GAS int*)src, (LAS int*)&tile[lofs], 0, 0);
#else
            tile[lofs] = *src;
#endif
        } else {
            tile[lofs] = 0.0f;             // zero padding
        }
    }
#if USE_ASYNC_COPY
#if __has_builtin(__builtin_amdgcn_s_wait_asynccnt)
    __builtin_amdgcn_s_wait_asynccnt(0);
#else
    asm volatile("s_wait_asynccnt 0" ::: "memory");
#endif
#endif
    __syncthreads();

    // ---- Per-thread static slot mapping ----
    // thread = (q, h): q = float4 slot 0..143 (elems e=4q..4q+3), h = pixel half.
    const int q = tid % NSLOT;
    const int h = tid / NSLOT;             // 0 or 1 -> pixels h*32 .. h*32+31
    const int pbase = h * HPIX;

    int b0 = gather_base(4 * q + 0) + pbase;
    int b1 = gather_base(4 * q + 1) + pbase;
    int b2 = gather_base(4 * q + 2) + pbase;
    int b3 = gather_base(4 * q + 3) + pbase;

    // Output float4 index: ((n*H + y)*W + x0 + pbase + i)*144 + q
    v4f* __restrict__ outp =
        reinterpret_cast<v4f*>(out) +
        (((size_t)(n * HH + y) * WW + x0 + pbase) * NSLOT + q);

    #pragma unroll 4
    for (int i = 0; i < HPIX; ++i) {
        v4f v;
        v.x = tile[b0 + i];                // LDS gathers (scrambled channel)
        v.y = tile[b1 + i];
        v.z = tile[b2 + i];
        v.w = tile[b3 + i];
        // Streaming output: non-temporal b128 store keeps input resident in L2.
        __builtin_nontemporal_store(v, outp + (size_t)i * NSLOT);
    }
}

extern "C" void kernel_launch(void* const* d_in, const int* in_sizes, int n_in,
                              void* d_out, int out_size, void* d_ws, size_t ws_size,
                              hipStream_t stream)
{
    (void)in_sizes; (void)n_in; (void)out_size; (void)d_ws; (void)ws_size;
    const float* x = (const float*)d_in[0];
    float* out = (float*)d_out;

    dim3 grid(WW / TX, HH, 8);   // (2, 128, 8) = 2048 workgroups
    dim3 block(NTHR, 1, 1);      // 288 threads = 9 waves (wave32)
    convchunk_gather_kernel<<<grid, block, 0, stream>>>(x, out);
}